// Backlash_86870008529012
// MI455X (gfx1250) — compile-verified
//
#include <hip/hip_runtime.h>
#include <hip/hip_bf16.h>
#include <stdint.h>

// Backlash scan: out_t = f(out_{t-1}, x_t, w_t) -- sequential in T, parallel in B.
// f1 = _h((m_lo*x + m_lo*c_lo - prev)/m_lo), f2 = _h((prev - m_up*x - m_up*c_up)/m_up)
// _h(s) = (s>0) ? 0 : 1. With f1,f2 in {0,1}:
//   out = prev          if f1==0 && f2==0
//       = U  (=m_up*(x+c_up))   if f1==0 && f2==1
//       = L  (=m_lo*(x+c_lo))   if f1==1 && f2==0
//       = L+U                   if f1==1 && f2==1
// c1 := (f1==0) <=> (L-prev)*m_lo > 0 ;  c2 := (f2==0) <=> (U-prev)*m_up < 0.
// Chain per step: sub -> mul -> cmp -> cndmask -> cndmask  (5 dependent VALU).

typedef unsigned uint4v __attribute__((ext_vector_type(4)));
typedef unsigned uint8v __attribute__((ext_vector_type(8)));

__device__ __forceinline__ float backlash_step(float prev, float xt,
                                               float m_lo, float m_up,
                                               float c_lo, float c_up) {
  float p1 = m_lo * xt;
  float p2 = m_lo * c_lo;
  float p3 = m_up * xt;
  float p4 = m_up * c_up;
  float L   = p1 + p2;      // lower engage value
  float U   = p3 + p4;      // upper engage value
  float sum = L + U;        // both engaged (rare)
  float a1 = L - prev;
  float a2 = U - prev;
  bool c1 = (a1 * m_lo) > 0.0f;  // f1 == 0
  bool c2 = (a2 * m_up) < 0.0f;  // f2 == 0
  float t = c2 ? prev : U;
  float e = c2 ? L    : sum;
  return c1 ? t : e;
}

// --- Tensor Data Mover: one 2D tiled DMA (global -> LDS) per descriptor. ---
// Group1 constant builder. data_size=4B (code 2), pad_enable, pad_interval code
// (2^(code+1) dwords), pad_amount code (code+1 dwords appended per interval).
__device__ __forceinline__ uint8v tdm_group1(unsigned tensor_dim0, unsigned tensor_dim1,
                                             unsigned tile_dim0, unsigned tile_dim1,
                                             unsigned d0_stride,
                                             unsigned pad_interval_code, unsigned pad_amount_code) {
  uint8v g;
  g[0] = (2u << 16) | (1u << 20) | (pad_interval_code << 22) | (pad_amount_code << 25);
  g[1] = (tensor_dim0 & 0xFFFFu) << 16;            // tensor_dim0[15:0]  @ bits 63:48
  g[2] = (tensor_dim0 >> 16) | ((tensor_dim1 & 0xFFFFu) << 16); // dim0[31:16], dim1[15:0]
  g[3] = (tensor_dim1 >> 16) | (tile_dim0 << 16);  // dim1[31:16], tile_dim0 @ 127:112
  g[4] = tile_dim1;                                 // tile_dim1 @ 143:128, tile_dim2=0
  g[5] = d0_stride;                                 // tensor_dim0_stride[31:0]
  g[6] = 0u;                                        // d0_stride[47:32]=0, d1_stride[15:0]=0
  g[7] = 0u;                                        // d1_stride[47:16]=0 (2D tile)
  return g;
}

__device__ __forceinline__ void tdm_load(unsigned lds_addr, uint64_t gaddr, uint8v g1) {
  uint4v g0;
  g0[0] = 1u;                                       // count=1, user mode, no gather
  g0[1] = lds_addr;                                 // LDS byte address
  g0[2] = (unsigned)gaddr;                          // global_addr[31:0]
  g0[3] = (unsigned)(gaddr >> 32) | 0x80000000u;    // global_addr[56:32] | type=2
  asm volatile("tensor_load_to_lds %0, %1" :: "s"(g0), "s"(g1) : "memory");
}

template <int T, int TC>
__global__ __launch_bounds__(32)
void backlash_scan_tdm(const float* __restrict__ x,
                       const float* __restrict__ out_prev,
                       const float* __restrict__ w,
                       float* __restrict__ out, int B) {
  constexpr int XP = TC + 4;       // x row pitch (dwords) == TDM pad: 32dw + 4dw
  constexpr int WP = 4 * TC + 4;   // w row pitch (dwords) == TDM pad: 128dw + 4dw
  constexpr int NT = T / TC;

  __shared__ float xs[2][32 * XP]; //  9 KB, double-buffered 32 x TC tile
  __shared__ float ws[2][32 * WP]; // 33 KB, double-buffered 32 x 4TC tile

  const int lane = threadIdx.x;    // one batch row per lane
  const int b0   = blockIdx.x * 32;
  const int b    = b0 + lane;

  const unsigned xlds[2] = { (unsigned)(uintptr_t)(&xs[0][0]),
                             (unsigned)(uintptr_t)(&xs[1][0]) };
  const unsigned wlds[2] = { (unsigned)(uintptr_t)(&ws[0][0]),
                             (unsigned)(uintptr_t)(&ws[1][0]) };

  // Loop-invariant descriptor group1 (stays in SGPRs).
  const uint8v g1x = tdm_group1(/*dim0=*/T,     /*dim1=*/(unsigned)B,
                                /*tile0=*/TC,   /*tile1=*/32,
                                /*stride=*/T,   /*pad_int=*/4, /*pad_amt=*/3);
  const uint8v g1w = tdm_group1(/*dim0=*/4 * T, /*dim1=*/(unsigned)B,
                                /*tile0=*/4 * TC, /*tile1=*/32,
                                /*stride=*/4 * T, /*pad_int=*/6, /*pad_amt=*/3);

  // Tile-start global addresses (advance per tile).
  uint64_t gx = (uint64_t)(uintptr_t)x + (uint64_t)b0 * T * 4u;
  uint64_t gw = (uint64_t)(uintptr_t)w + (uint64_t)b0 * T * 16u;

  float prev = out_prev[b];
  float* orow_base = out + (size_t)b * T;

  // Prologue: tile 0 in flight (2 tensor DMAs).
  tdm_load(xlds[0], gx, g1x);
  tdm_load(wlds[0], gw, g1w);

  for (int k = 0; k < NT; ++k) {
    const int buf = k & 1;
    if (k + 1 < NT) {
      gx += TC * 4u;                                  // next tile start
      gw += TC * 16u;
      tdm_load(xlds[buf ^ 1], gx, g1x);
      tdm_load(wlds[buf ^ 1], gw, g1w);
      asm volatile("s_wait_tensorcnt 0x2" ::: "memory"); // tile k resident, k+1 in flight
    } else {
      asm volatile("s_wait_tensorcnt 0x0" ::: "memory");
    }

    const float* xrow = &xs[buf][lane * XP];
    const float* wrow = &ws[buf][lane * WP];
    float* orow = orow_base + k * TC;

#pragma unroll 2
    for (int u = 0; u < TC / 4; ++u) {
      const float4 x4 = *(const float4*)(xrow + u * 4);
      float4 o4;
      {
        const float4 w4 = *(const float4*)(wrow + (u * 4 + 0) * 4);
        prev = backlash_step(prev, x4.x, w4.x, w4.y, w4.z, w4.w); o4.x = prev;
      }
      {
        const float4 w4 = *(const float4*)(wrow + (u * 4 + 1) * 4);
        prev = backlash_step(prev, x4.y, w4.x, w4.y, w4.z, w4.w); o4.y = prev;
      }
      {
        const float4 w4 = *(const float4*)(wrow + (u * 4 + 2) * 4);
        prev = backlash_step(prev, x4.z, w4.x, w4.y, w4.z, w4.w); o4.z = prev;
      }
      {
        const float4 w4 = *(const float4*)(wrow + (u * 4 + 3) * 4);
        prev = backlash_step(prev, x4.w, w4.x, w4.y, w4.z, w4.w); o4.w = prev;
      }
      *(float4*)(orow + u * 4) = o4; // aligned 16B store
    }
  }
}

// Fallback for unexpected shapes: plain per-lane scan straight from global.
__global__ void backlash_scan_generic(const float* __restrict__ x,
                                      const float* __restrict__ out_prev,
                                      const float* __restrict__ w,
                                      float* __restrict__ out, int B, int T) {
  int b = blockIdx.x * blockDim.x + threadIdx.x;
  if (b >= B) return;
  float prev = out_prev[b];
  const float* xr = x + (size_t)b * T;
  const float* wr = w + (size_t)b * T * 4;
  float* orr = out + (size_t)b * T;
  for (int t = 0; t < T; ++t) {
    float4 w4 = *(const float4*)(wr + t * 4);
    prev = backlash_step(prev, xr[t], w4.x, w4.y, w4.z, w4.w);
    orr[t] = prev;
  }
}

extern "C" void kernel_launch(void* const* d_in, const int* in_sizes, int n_in,
                              void* d_out, int out_size, void* d_ws, size_t ws_size,
                              hipStream_t stream) {
  const float* x  = (const float*)d_in[0];   // (B, T, 1)
  const float* op = (const float*)d_in[1];   // (B, 1, 1)
  const float* w  = (const float*)d_in[2];   // (B, T, 4)
  float* out = (float*)d_out;                // (B, T, 1)

  const int B = in_sizes[1];
  const int T = (B > 0) ? in_sizes[0] / B : 0;

  if (T == 2048 && B > 0 && (B % 32) == 0) {
    backlash_scan_tdm<2048, 32><<<B / 32, 32, 0, stream>>>(x, op, w, out, B);
  } else if (B > 0 && T > 0) {
    const int thr = 256;
    backlash_scan_generic<<<(B + thr - 1) / thr, thr, 0, stream>>>(x, op, w, out, B, T);
  }
}